// Glycolysis_31602369364720
// MI455X (gfx1250) — compile-verified
//
#include <hip/hip_runtime.h>

// Glycolysis ODE RHS: conc[28,B] f32 -> ddt[27,B] f32.
// HBM-bound streaming: 220 B/member -> ~9.4 us floor at 23.3 TB/s.
// This version: 2 columns per thread (float2 / b64 vmem), so per 440 bytes we
// issue 28 loads + 27 stores and let the backend pack the f32 math into
// v_pk_fma_f32/v_pk_mul_f32/v_pk_add_f32 (CDNA5 VOP3P). Non-temporal policy on
// all traffic (footprint > 192 MB L2, zero reuse). v_rcp_f32 for reciprocals.

#define DEV __device__ __forceinline__

typedef float v2f __attribute__((ext_vector_type(2)));

DEV float rcpx(float x) { return __builtin_amdgcn_rcpf(x); }
DEV v2f   rcpx(v2f x) {
    v2f r;
    r.x = __builtin_amdgcn_rcpf(x.x);
    r.y = __builtin_amdgcn_rcpf(x.y);
    return r;
}

// cc[28] concentrations -> dd[27] derivatives. T = float or v2f.
template <typename T>
DEV void glyco_eval(const T* __restrict__ cc,
                    const float* __restrict__ vmax,
                    const float* __restrict__ km,
                    const float* __restrict__ keq,
                    T* __restrict__ dd)
{
    const T Glci = cc[0],  Glco = cc[1],  ATP  = cc[2],  ADP  = cc[3];
    const T G6P  = cc[4],  T6P  = cc[5],  G1P  = cc[6],  F6P  = cc[7];
    const T PI_  = cc[8],  TRE  = cc[9],  F16BP= cc[10], AMP  = cc[11];
    const T DHAP = cc[12], G3P  = cc[13], NAD  = cc[14], GLYC = cc[15];
    const T BPG  = cc[16], NADH = cc[17], P3G  = cc[18], P2G  = cc[19];
    const T PEP  = cc[20], PYR  = cc[21], ETOH = cc[22], ACE  = cc[23];
    const T IMP  = cc[24], INO  = cc[25], HYP  = cc[26], GAP  = cc[27];

    #define SAT(s, k)      ((s) * rcpx((s) + (k)))          // s/(k+s)
    #define INH(x, k)      ((k) * rcpx((x) + (k)))          // k/(k+x)
    #define MM(idx, s)     (vmax[idx] * SAT((s), km[idx]))
    #define REV(idx, s, p) (vmax[idx] * ((s) - (p) * rcpx(keq[idx])) \
                            * rcpx(km[idx] + (s) + (p)))

    const float UDP_GLC = 0.07f;

    const T f0  = REV(0, Glco, Glci);                                            // v_GLT
    const T f1  = vmax[1]*SAT(ATP,km[1])*SAT(Glci,km[1])
                  *INH(ADP,km[1])*INH(G6P,km[1])*INH(T6P,km[1]);                 // v_GLK
    const T f2  = REV(2, G1P, G6P);                                              // v_PGM1
    const T f3  = vmax[3]*SAT(G6P,km[3])*(UDP_GLC*rcpx(km[3]+UDP_GLC))
                  *INH(F6P,km[3])*INH(PI_,km[3]);                                // v_TPS1
    const T f4  = vmax[4]*SAT(T6P,km[4])*INH(PI_,km[4]);                         // v_TPS2
    const T f5  = MM(5, TRE);                                                    // v_NTH1
    const T f6  = vmax[6]*SAT(F6P,km[6])*INH(ATP,km[6]);                         // v_PGI
    const T f7  = vmax[7]*SAT(ATP,km[7])*SAT(Glci,km[7])
                  *INH(F16BP,km[7])*(1.0f + AMP*rcpx(km[7]));                    // v_PFK
    const T f8  = vmax[8]*SAT(F16BP,km[8])*INH(F6P,km[8])*INH(DHAP,km[8]);       // v_ALD
    const T f9  = REV(9, G3P, G3P);                                              // v_TPI
    const T i10 = INH(G3P,km[10]);
    const T f10 = vmax[10]*SAT(DHAP,km[10])*SAT(NAD,km[10])
                  *i10*i10*INH(F16BP,km[10])*INH(ATP,km[10])*INH(ADP,km[10]);    // v_G3PDH
    const T f11 = vmax[11]*SAT(G3P,km[11])*INH(PI_,km[11]);                      // v_HOR2
    const T f12 = MM(12, GLYC);                                                  // v_GlycT
    const T f13 = vmax[13]*SAT(G3P,km[13])*SAT(NAD,km[13])*SAT(PI_,km[13])
                  *INH(BPG,km[13])*INH(NADH,km[13]);                             // v_GAPDH
    const T f14 = vmax[14]*(BPG*ADP - P3G*ATP*rcpx(keq[14]))
                  * rcpx(km[14]+BPG+P3G) * rcpx(km[14]+ADP+ATP);                 // v_PGK
    const T f15 = REV(15, P3G, P2G);                                             // v_PGM
    const T f16 = REV(16, P2G, PEP);                                             // v_ENO
    const T f17 = vmax[17]*SAT(PEP,km[17])*SAT(ADP,km[17])
                  *INH(ATP,km[17])*(1.0f + F16BP*rcpx(km[17]));                  // v_PYK
    const T f18 = vmax[18]*SAT(PYR,km[18])*INH(PI_,km[18]);                      // v_PDC
    const T f19 = vmax[19]*(NAD*ETOH - ACE*NADH*rcpx(keq[19]))
                  * rcpx(km[19]+NAD+NADH) * rcpx(km[19]+ETOH+ACE);               // v_ADH
    const T f20 = MM(20, ETOH);                                                  // v_EtohT
    const T f21 = vmax[21]*SAT(PI_,km[21])*SAT(ADP,km[21]);                      // v_ATPmito
    const T f22 = vmax[22]*SAT(ATP,km[22])*INH(ADP,km[22]);                      // v_ATPase
    const T adp2 = ADP*ADP, atpamp = ATP*AMP;
    const T f23 = vmax[23]*(adp2 - atpamp*rcpx(keq[23]))
                  * rcpx(km[23] + adp2 + atpamp);                                // v_ADK1
    const T f24 = MM(24, PI_);                                                   // v_vacPi
    const T f25 = vmax[25]*SAT(AMP,km[25])*INH(ATP,km[25])*INH(PI_,km[25]);      // v_Amd1
    const T f26 = MM(26, IMP);                                                   // v_Ade1312
    const T f27 = MM(27, IMP);                                                   // v_Isn1
    const T f28 = MM(28, INO);                                                   // v_Pnp1
    const T f29 = MM(29, HYP);                                                   // v_Hpt1
    const T f30 = MM(30, NADH);                                                  // v_NADHmito
    const T f31 = MM(31, G6P);                                                   // vsinkG6P
    const T f32_= MM(32, F6P);                                                   // vsinkF6P
    const T f33 = MM(33, GAP);                                                   // vsinkGAP
    const T f34 = MM(34, P3G);                                                   // vsinkP3G
    const T f35 = MM(35, PEP);                                                   // vsinkPEP
    const T f36 = MM(36, PYR);                                                   // vsinkPYR
    const T f37 = MM(37, ACE);                                                   // vsinkACE

    #undef SAT
    #undef INH
    #undef MM
    #undef REV

    dd[0]  = -f1 + f0 + 2.0f*f5;                                  // GLCi
    dd[1]  =  f1 - f6 + f31 + f2 - f3;                            // G6P
    dd[2]  = -f2 - 0.00031f;                                      // G1P
    dd[3]  =  f3 - f4;                                            // T6P
    dd[4]  =  f4 - f5;                                            // TRE
    dd[5]  = -f7 + f6 + f32_;                                     // F6P
    dd[6]  =  f7 - f8;                                            // F16BP
    dd[7]  =  f8 - f13 + f9 + f33;                                // GAP
    dd[8]  =  f8 - f9 - f10;                                      // DHAP
    dd[9]  =  f10 - f11;                                          // G3P
    dd[10] =  f11 - f12;                                          // GLYCEROL
    dd[11] =  f13 - f14;                                          // BPG
    dd[12] =  f14 - f15 + f34;                                    // P3G
    dd[13] =  f15 - f16;                                          // P2G
    dd[14] =  f16 - f17 + f35;                                    // PEP
    dd[15] =  f17 - f18 + f36;                                    // PYR
    dd[16] =  f18 - f19 + f37;                                    // ACE
    dd[17] =  f19 - f20;                                          // ETOH
    dd[18] =  f23 - f1 - f22 - f7 + f14 + f17 - f3 + f21;         // ATP
    dd[19] = -2.0f*f23 + f1 + f22 + f7 - f14 - f17 + f3 - f21;    // ADP
    dd[20] =  f23 - f25 + f26;                                    // AMP
    dd[21] = -f13 + f22 + f11 + 2.0f*f3 + f4 - f21 + f27 - f28
             + f24 - f31 - f32_ - f33 - f34 - f35;                // PI
    dd[22] =  f25 - f26 + f29 - f27;                              // IMP
    dd[23] =  f27 - f28;                                          // INO
    dd[24] =  f28 - f29;                                          // HYP
    dd[25] =  f10 - f13 + f19 + f30;                              // NAD
    dd[26] = -f10 + f13 - f19 - f30;                              // NADH
}

// Main path: 2 columns per thread, b64 vmem, packed-f32 math.
__global__ __launch_bounds__(256) void glyco_rhs_v2(
    const float* __restrict__ conc, const float* __restrict__ vmax,
    const float* __restrict__ km,   const float* __restrict__ keq,
    float* __restrict__ out, int B)
{
    const int p = blockIdx.x * blockDim.x + threadIdx.x;   // pair index
    if (p >= (B >> 1)) return;

    v2f cc[28];
    #pragma unroll
    for (int m = 0; m < 28; ++m)
        cc[m] = __builtin_nontemporal_load(((const v2f*)(conc + (size_t)m * B)) + p);

    v2f dd[27];
    glyco_eval<v2f>(cc, vmax, km, keq, dd);

    #pragma unroll
    for (int r = 0; r < 27; ++r)
        __builtin_nontemporal_store(dd[r], ((v2f*)(out + (size_t)r * B)) + p);
}

// Tail path for odd B: scalar, covers columns [i0, B).
__global__ __launch_bounds__(64) void glyco_rhs_tail(
    const float* __restrict__ conc, const float* __restrict__ vmax,
    const float* __restrict__ km,   const float* __restrict__ keq,
    float* __restrict__ out, int B, int i0)
{
    const int i = i0 + blockIdx.x * blockDim.x + threadIdx.x;
    if (i >= B) return;

    float cc[28];
    #pragma unroll
    for (int m = 0; m < 28; ++m)
        cc[m] = __builtin_nontemporal_load(conc + (size_t)m * B + i);

    float dd[27];
    glyco_eval<float>(cc, vmax, km, keq, dd);

    #pragma unroll
    for (int r = 0; r < 27; ++r)
        __builtin_nontemporal_store(dd[r], out + (size_t)r * B + i);
}

extern "C" void kernel_launch(void* const* d_in, const int* in_sizes, int n_in,
                              void* d_out, int out_size, void* d_ws, size_t ws_size,
                              hipStream_t stream) {
    const float* conc = (const float*)d_in[0];
    const float* vmax = (const float*)d_in[1];
    const float* km   = (const float*)d_in[2];
    const float* keq  = (const float*)d_in[3];
    float* out = (float*)d_out;

    const int B = in_sizes[0] / 28;
    const int npair = B >> 1;

    if (npair > 0) {
        const int threads = 256;
        const int blocks = (npair + threads - 1) / threads;
        glyco_rhs_v2<<<blocks, threads, 0, stream>>>(conc, vmax, km, keq, out, B);
    }
    if (B & 1) {
        glyco_rhs_tail<<<1, 64, 0, stream>>>(conc, vmax, km, keq, out, B, B & ~1);
    }
}